// TwoStageClassificationLoss_80341658239298
// MI455X (gfx1250) — compile-verified
//
#include <hip/hip_runtime.h>
#include <hip/hip_bf16.h>

typedef float v2f __attribute__((ext_vector_type(2)));
typedef float v8f __attribute__((ext_vector_type(8)));

#define LOG2E_CLAMP  (-144.269504089f)   // -100 / ln(2)
#define NEG_LN2      (-0.693147180559945f)
#define DELTA        0.5f

// BCE contribution in log2 domain (positive-accumulated; scale by -ln2 at finalize):
//   contrib = t*max(log2 p, C) + (1-t)*max(log2(1-p), C),  C = -100/ln2
__device__ __forceinline__ void bce_elem(float p, float t, float& acc) {
    p = fminf(fmaxf(p, 0.0f), 1.0f);               // v_med3
    t = fminf(fmaxf(t, 0.0f), 1.0f);
    float l2p = fmaxf(__log2f(p),        LOG2E_CLAMP);   // log2(0) = -inf -> clamp
    float l2q = fmaxf(__log2f(1.0f - p), LOG2E_CLAMP);
    acc += fmaf(t, l2p - l2q, l2q);                // t*l2p + (1-t)*l2q
}

// duration focal smooth-L1; ALPHA folded into the class weight:
//   weights {0:1,1:4,2:3,3:2} * 0.25  ->  w' = (r==0) ? 0.25 : 1.25 - 0.25*r
__device__ __forceinline__ void dur_elem(float yp, float yt, float& acc) {
    float d = fabsf(yp - yt);
    float r = fminf(fmaxf(rintf(yt), 0.0f), 3.0f);       // round-half-even, clamp [0,3]
    float w = (r < 0.5f) ? 0.25f : fmaf(-0.25f, r, 1.25f);
    float f = fminf(d + d, 1.0f);                        // clip(d/DELTA, 0, 1)
    float focal = f * f;                                 // gamma = 2
    float base  = (d < DELTA) ? (d * d) : (d - 0.25f);   // smooth-L1, beta = 0.5
    acc = fmaf(focal * base, w, acc);
}

// Exact f32 wave32 reduction of (bceA, durA) via two V_WMMA_F32_16X16X4_F32.
// Lanes 0..15 end with res[0] = wave sum of bce, res[1] = wave sum of dur.
__device__ __forceinline__ v2f wave_reduce_wmma(float bceA, float durA, unsigned lane) {
    v2f a;  a[0] = bceA;  a[1] = durA;              // A[16x4]: lane L -> row L%16
    v2f b1;                                          // B col0 = (1,0,1,0), col1 = (0,1,0,1)
    b1[0] = ((lane & 15u) == 0u) ? 1.0f : 0.0f;
    b1[1] = ((lane & 15u) == 1u) ? 1.0f : 0.0f;
    v8f c = {0.f, 0.f, 0.f, 0.f, 0.f, 0.f, 0.f, 0.f};
    // D1[m][0] = bce[m]+bce[m+16]; D1[m][1] = dur[m]+dur[m+16]
    v8f d1 = __builtin_amdgcn_wmma_f32_16x16x4_f32(false, a, false, b1,
                                                   (short)0, c, false, false);
    v2f a2;                                          // quarter row-sums as new A matrix
    a2[0] = (d1[0] + d1[1]) + (d1[2] + d1[3]);
    a2[1] = (d1[4] + d1[5]) + (d1[6] + d1[7]);
    v2f b2;  b2[0] = 1.0f;  b2[1] = 1.0f;            // B2 = ones -> D2[m][n] = colsum_m(D1)
    v8f d2 = __builtin_amdgcn_wmma_f32_16x16x4_f32(false, a2, false, b2,
                                                   (short)0, c, false, false);
    v2f res;  res[0] = d2[0];  res[1] = d2[1];       // rows 0/1, valid on lanes 0..15
    return res;
}

__global__ void __launch_bounds__(256)
loss_main_kernel(const float4* __restrict__ bp, const float4* __restrict__ bt,
                 const float4* __restrict__ sp, const float4* __restrict__ st,
                 float* __restrict__ acc, int n4) {
    float bceA = 0.0f, durA = 0.0f;
    const int stride = gridDim.x * blockDim.x;
    const int gid = blockIdx.x * blockDim.x + threadIdx.x;

    int i = gid;
    // unroll x2: 8 b128 loads in flight per iteration
    for (; i + stride < n4; i += 2 * stride) {
        float4 p0  = bp[i];          float4 p1  = bp[i + stride];
        float4 t0  = bt[i];          float4 t1  = bt[i + stride];
        float4 yp0 = sp[i];          float4 yp1 = sp[i + stride];
        float4 yt0 = st[i];          float4 yt1 = st[i + stride];
        bce_elem(p0.x, t0.x, bceA);  bce_elem(p0.y, t0.y, bceA);
        bce_elem(p0.z, t0.z, bceA);  bce_elem(p0.w, t0.w, bceA);
        bce_elem(p1.x, t1.x, bceA);  bce_elem(p1.y, t1.y, bceA);
        bce_elem(p1.z, t1.z, bceA);  bce_elem(p1.w, t1.w, bceA);
        dur_elem(yp0.x, yt0.x, durA); dur_elem(yp0.y, yt0.y, durA);
        dur_elem(yp0.z, yt0.z, durA); dur_elem(yp0.w, yt0.w, durA);
        dur_elem(yp1.x, yt1.x, durA); dur_elem(yp1.y, yt1.y, durA);
        dur_elem(yp1.z, yt1.z, durA); dur_elem(yp1.w, yt1.w, durA);
    }
    if (i < n4) {
        float4 p = bp[i], t = bt[i], yp = sp[i], yt = st[i];
        bce_elem(p.x, t.x, bceA);  bce_elem(p.y, t.y, bceA);
        bce_elem(p.z, t.z, bceA);  bce_elem(p.w, t.w, bceA);
        dur_elem(yp.x, yt.x, durA); dur_elem(yp.y, yt.y, durA);
        dur_elem(yp.z, yt.z, durA); dur_elem(yp.w, yt.w, durA);
    }

    // Full-EXEC wave32 reduction via WMMA
    const unsigned lane = threadIdx.x & 31u;
    v2f sums = wave_reduce_wmma(bceA, durA, lane);

    __shared__ float sAcc[2];
    if (threadIdx.x < 2) sAcc[threadIdx.x] = 0.0f;
    __syncthreads();
    if (lane == 0) {
        atomicAdd(&sAcc[0], sums[0]);   // ds_add_f32
        atomicAdd(&sAcc[1], sums[1]);
    }
    __syncthreads();
    if (threadIdx.x < 2) {
        __hip_atomic_fetch_add(&acc[threadIdx.x], sAcc[threadIdx.x],
                               __ATOMIC_RELAXED, __HIP_MEMORY_SCOPE_AGENT);
    }
}

__global__ void zero_acc_kernel(float* __restrict__ acc) {
    if (threadIdx.x < 2) acc[threadIdx.x] = 0.0f;
}

// Adds the <=3 tail elements (n % 4), then converts sums to means & writes outputs.
__global__ void finalize_kernel(const float* __restrict__ acc,
                                const float* __restrict__ bp, const float* __restrict__ bt,
                                const float* __restrict__ sp, const float* __restrict__ st,
                                float* __restrict__ out, int tailStart, int n, float invN) {
    if (threadIdx.x == 0) {
        float bceT = 0.0f, durT = 0.0f;
        for (int k = tailStart; k < n; ++k) {
            bce_elem(bp[k], bt[k], bceT);
            dur_elem(sp[k], st[k], durT);
        }
        float bce = (acc[0] + bceT) * (NEG_LN2 * invN);  // back to ln domain, mean, negate
        float dur = (acc[1] + durT) * invN;
        out[0] = bce + dur;   // total
        out[1] = bce;
        out[2] = dur;
    }
}

extern "C" void kernel_launch(void* const* d_in, const int* in_sizes, int n_in,
                              void* d_out, int out_size, void* d_ws, size_t ws_size,
                              hipStream_t stream) {
    const float* bp = (const float*)d_in[0];   // break_predictions
    const float* bt = (const float*)d_in[1];   // break_targets
    const float* sp = (const float*)d_in[2];   // stage2_predictions
    const float* st = (const float*)d_in[3];   // stage2_targets
    float* out = (float*)d_out;
    float* acc = (float*)d_ws;                 // acc[0] = bce log2-sum, acc[1] = dur sum

    const int n  = in_sizes[0];                // 64 * 131072 = 8388608
    const int n4 = n >> 2;

    zero_acc_kernel<<<1, 32, 0, stream>>>(acc);

    const int threads = 256;                   // 8 wave32 per block
    int blocks = (n4 + threads - 1) / threads;
    if (blocks > 4096) blocks = 4096;          // 2 float4 per thread at full size
    if (blocks < 1) blocks = 1;

    loss_main_kernel<<<blocks, threads, 0, stream>>>(
        (const float4*)bp, (const float4*)bt,
        (const float4*)sp, (const float4*)st, acc, n4);

    finalize_kernel<<<1, 32, 0, stream>>>(acc, bp, bt, sp, st, out,
                                          n4 << 2, n, 1.0f / (float)n);
}